// RunningReturn_25254407701164
// MI455X (gfx1250) — compile-verified
//
#include <hip/hip_runtime.h>
#include <stdint.h>

// Problem constants (match reference)
#define TT      2048
#define BB      8192
#define GAMMA_F 0.99f
#define NCH     32          // chunks along T
#define CLEN    64          // TT / NCH
#define CPB     1024        // columns per block in pass 1
#define NCG     (BB / CPB)  // 8 column groups
#define STEP    4           // rows per LDS buffer
#define NSTAT   7           // S1, SS, SP, SPs, SP2, sEnd, PEnd

// ---------------- CDNA5 async global->LDS helpers ----------------

template <int N>
__device__ __forceinline__ void async_wait_le() {
#if __has_builtin(__builtin_amdgcn_s_wait_asynccnt)
  __builtin_amdgcn_s_wait_asynccnt(N);
#else
  asm volatile("s_wait_asynccnt %0" :: "i"(N) : "memory");
#endif
}

__device__ __forceinline__ void ds_reads_drain() {
#if __has_builtin(__builtin_amdgcn_s_wait_dscnt)
  __builtin_amdgcn_s_wait_dscnt(0);
#else
  asm volatile("s_wait_dscnt 0" ::: "memory");
#endif
}

typedef int v4i __attribute__((ext_vector_type(4)));
typedef __attribute__((address_space(1))) v4i* g4p;   // global v4i32*
typedef __attribute__((address_space(3))) v4i* l4p;   // LDS v4i32*

__device__ __forceinline__ void async_copy16(const float* gsrc, float* ldst) {
#if __has_builtin(__builtin_amdgcn_global_load_async_to_lds_b128)
  __builtin_amdgcn_global_load_async_to_lds_b128(
      (g4p)(uintptr_t)gsrc, (l4p)(uint32_t)(uintptr_t)ldst, 0, 0);
#else
  unsigned           lofs = (unsigned)(uintptr_t)ldst;           // low 32 bits = LDS byte offset
  unsigned long long gadr = (unsigned long long)(uintptr_t)gsrc; // global address
  asm volatile("global_load_async_to_lds_b128 %0, %1, off"
               :: "v"(lofs), "v"(gadr) : "memory");
#endif
}

// ---------------- Pass 1: chunked scan, 4 columns per thread ----------------
// Each block: one chunk (64 timesteps) x 1024 columns. Double-buffered async
// LDS staging of x and dones rows; per-chunk affine-scan statistics out.

__global__ __launch_bounds__(256)
void rr_pass1_scan(const float* __restrict__ x, const float* __restrict__ dones,
                   float* __restrict__ stats) {
  __shared__ float xs[2][STEP][CPB];
  __shared__ float dsb[2][STEP][CPB];

  const int tid  = threadIdx.x;
  const int cgrp = blockIdx.x % NCG;
  const int c    = blockIdx.x / NCG;      // chunk id
  const int col  = cgrp * CPB + tid * 4;  // 4 consecutive columns
  const int t0   = c * CLEN;

  auto issue = [&](int bi, int t) {
#pragma unroll
    for (int r = 0; r < STEP; ++r) {
      const float* gx = x + (size_t)(t + r) * BB + col;
      int dr = t + r - 1; if (dr < 0) dr = 0;           // t==0 handled in compute
      const float* gd = dones + (size_t)dr * BB + col;
      async_copy16(gx, &xs[bi][r][tid * 4]);
      async_copy16(gd, &dsb[bi][r][tid * 4]);
    }
  };

  issue(0, t0);
  issue(1, t0 + STEP);

  float P[4], s[4], S1[4], SS[4], SP[4], SPs[4], SP2[4];
#pragma unroll
  for (int k = 0; k < 4; ++k) {
    P[k] = 1.0f; s[k] = 0.0f;
    S1[k] = SS[k] = SP[k] = SPs[k] = SP2[k] = 0.0f;
  }

  for (int base = 0; base < CLEN; base += STEP) {
    const int bi = (base / STEP) & 1;
    if (base + STEP < CLEN) async_wait_le<2 * STEP>();  // this buffer done, next in flight
    else                    async_wait_le<0>();

#pragma unroll
    for (int r = 0; r < STEP; ++r) {
      const int t = t0 + base + r;
      float4 xv = *(const float4*)&xs[bi][r][tid * 4];
      float4 dv = *(const float4*)&dsb[bi][r][tid * 4];
      if (t == 0) { dv.x = dv.y = dv.z = dv.w = 0.0f; }  // d_prev[0] = 0
      const float xa[4] = {xv.x, xv.y, xv.z, xv.w};
      const float da[4] = {dv.x, dv.y, dv.z, dv.w};
#pragma unroll
      for (int k = 0; k < 4; ++k) {
        const float a = GAMMA_F * (1.0f - da[k]);   // exactly GAMMA or 0
        P[k] *= a;
        s[k]  = fmaf(a, s[k], xa[k]);
        S1[k] += s[k];
        SS[k]  = fmaf(s[k], s[k], SS[k]);
        SP[k] += P[k];
        SPs[k] = fmaf(P[k], s[k], SPs[k]);
        SP2[k] = fmaf(P[k], P[k], SP2[k]);
      }
    }
    if (base + 2 * STEP < CLEN) {
      // WAR guard: ensure our DS reads of this buffer completed before the
      // async engine is allowed to overwrite it.
      ds_reads_drain();
      issue(bi, t0 + base + 2 * STEP);
    }
  }

  // stats layout: stats[(c*NSTAT + j)*BB + col]
  float* sb = stats + (size_t)c * NSTAT * BB + col;
  *(float4*)(sb + 0 * BB) = make_float4(S1[0], S1[1], S1[2], S1[3]);
  *(float4*)(sb + 1 * BB) = make_float4(SS[0], SS[1], SS[2], SS[3]);
  *(float4*)(sb + 2 * BB) = make_float4(SP[0], SP[1], SP[2], SP[3]);
  *(float4*)(sb + 3 * BB) = make_float4(SPs[0], SPs[1], SPs[2], SPs[3]);
  *(float4*)(sb + 4 * BB) = make_float4(SP2[0], SP2[1], SP2[2], SP2[3]);
  *(float4*)(sb + 5 * BB) = make_float4(s[0], s[1], s[2], s[3]);
  *(float4*)(sb + 6 * BB) = make_float4(P[0], P[1], P[2], P[3]);
}

// ---------------- Pass 2: stitch chunks, reduce, Welford, emit scale --------

__global__ __launch_bounds__(1024)
void rr_pass2_combine(const float* __restrict__ stats, const float* __restrict__ ret0,
                      const float* __restrict__ mean_in, const float* __restrict__ var_in,
                      const float* __restrict__ count_in, float* __restrict__ ws) {
  __shared__ double sh_s[1024];
  __shared__ double sh_q[1024];
  const int tid = threadIdx.x;

  double accS = 0.0, accQ = 0.0;
  for (int cc = 0; cc < BB / 1024; ++cc) {        // 8 columns per thread
    const int col = cc * 1024 + tid;
    double r = (double)ret0[col];
    for (int c = 0; c < NCH; ++c) {
      const float* st = stats + (size_t)c * NSTAT * BB + col;
      const double S1  = st[0 * BB];
      const double SSv = st[1 * BB];
      const double SP  = st[2 * BB];
      const double SPs = st[3 * BB];
      const double SP2 = st[4 * BB];
      const double sE  = st[5 * BB];
      const double PE  = st[6 * BB];
      accS += S1 + r * SP;
      accQ += SSv + 2.0 * r * SPs + r * r * SP2;
      r = sE + PE * r;
    }
  }
  sh_s[tid] = accS; sh_q[tid] = accQ;
  __syncthreads();
  for (int off = 512; off > 0; off >>= 1) {
    if (tid < off) { sh_s[tid] += sh_s[tid + off]; sh_q[tid] += sh_q[tid + off]; }
    __syncthreads();
  }
  if (tid == 0) {
    const double n     = (double)TT * (double)BB;
    const double bmean = sh_s[0] / n;
    double       bvar  = sh_q[0] / n - bmean * bmean;
    if (bvar < 0.0) bvar = 0.0;

    const double m = (double)mean_in[0], v = (double)var_in[0], cnt = (double)count_in[0];
    double nvar;
    if (cnt == 0.0) {
      nvar = bvar;
    } else {
      const double delta = bmean - m;
      const double tot   = cnt + n;
      const double m2    = v * cnt + bvar * n + delta * delta * cnt * n / tot;
      nvar = m2 / tot;
    }
    const double ov = nvar > 1e-6 ? nvar : 1e-6;
    ws[0] = (float)(1.0 / sqrt(ov));
  }
}

// ---------------- Pass 3: out = clip(x * scale, -10, 10) --------------------

__global__ __launch_bounds__(256)
void rr_pass3_norm(const float* __restrict__ x, const float* __restrict__ ws,
                   float* __restrict__ out) {
  const float sc = ws[0];
  const size_t n4     = (size_t)TT * BB / 4;
  const size_t stride = (size_t)gridDim.x * blockDim.x;
  const float4* x4 = (const float4*)x;
  float4*       o4 = (float4*)out;
  for (size_t i = (size_t)blockIdx.x * blockDim.x + threadIdx.x; i < n4; i += stride) {
    float4 v = x4[i];
    v.x = fminf(10.0f, fmaxf(-10.0f, v.x * sc));
    v.y = fminf(10.0f, fmaxf(-10.0f, v.y * sc));
    v.z = fminf(10.0f, fmaxf(-10.0f, v.z * sc));
    v.w = fminf(10.0f, fmaxf(-10.0f, v.w * sc));
    o4[i] = v;
  }
}

// ---------------- Launch --------------------------------------------------

extern "C" void kernel_launch(void* const* d_in, const int* in_sizes, int n_in,
                              void* d_out, int out_size, void* d_ws, size_t ws_size,
                              hipStream_t stream) {
  const float* x     = (const float*)d_in[0];
  const float* dones = (const float*)d_in[1];
  const float* ret0  = (const float*)d_in[2];
  const float* mean  = (const float*)d_in[3];
  const float* var   = (const float*)d_in[4];
  const float* count = (const float*)d_in[5];
  float*       out   = (float*)d_out;
  float*       ws    = (float*)d_ws;

  // Reuse d_out as scratch for chunk stats (7 MB of 64 MB); pass 3 fully
  // overwrites d_out afterwards. Stream order serializes the passes.
  float* stats = out;

  rr_pass1_scan<<<NCH * NCG, 256, 0, stream>>>(x, dones, stats);
  rr_pass2_combine<<<1, 1024, 0, stream>>>(stats, ret0, mean, var, count, ws);
  rr_pass3_norm<<<4096, 256, 0, stream>>>(x, ws, out);
}